// Net_27436251087104
// MI455X (gfx1250) — compile-verified
//
#include <hip/hip_runtime.h>

#define NN 50000
#define NE 600000
#define EPS_BN 1e-5f

typedef __attribute__((ext_vector_type(2))) float v2f;
typedef __attribute__((ext_vector_type(8))) float v8f;

// ---------------- normalization precompute ----------------

__global__ void deg_init_k(float* __restrict__ deg, int n) {
    int i = blockIdx.x * blockDim.x + threadIdx.x;
    if (i < n) deg[i] = 1.0f;  // self-loop weight
}

__global__ void deg_edge_k(const int* __restrict__ dst, const float* __restrict__ w,
                           float* __restrict__ deg, int e) {
    int i = blockIdx.x * blockDim.x + threadIdx.x;
    if (i < e) atomicAdd(&deg[dst[i]], w[i]);
}

__global__ void dinv_k(const float* __restrict__ deg, float* __restrict__ dinv, int n) {
    int i = blockIdx.x * blockDim.x + threadIdx.x;
    if (i < n) {
        float d = deg[i];
        dinv[i] = d > 0.0f ? rsqrtf(d) : 0.0f;
    }
}

__global__ void norm_k(const int* __restrict__ src, const int* __restrict__ dst,
                       const float* __restrict__ w, const float* __restrict__ dinv,
                       float* __restrict__ norm, int e) {
    int i = blockIdx.x * blockDim.x + threadIdx.x;
    if (i < e) norm[i] = dinv[src[i]] * w[i] * dinv[dst[i]];
}

// ---------------- fp32 WMMA GEMM: Y[n,FOUT] = X[n,FIN] @ W[FIN,FOUT] ----------------
// One 16-row tile per block; each wave owns a 16-col tile. K stepped by 4 via
// V_WMMA_F32_16X16X4_F32. A staged in LDS (padded), B streamed from global (L2-resident).

template <int FIN, int FOUT>
__global__ __launch_bounds__(32 * (FOUT / 16))
void gemm_wmma_k(const float* __restrict__ X, const float* __restrict__ W,
                 float* __restrict__ Y, int n) {
    constexpr int WAVES = FOUT / 16;
    constexpr int LDSP  = FIN + 1;            // pad to break 16-way bank conflicts
    __shared__ float xs[16 * LDSP];

    const int rowBase = blockIdx.x * 16;
    const int tid     = threadIdx.x;

    // stage 16 rows of X into LDS
    for (int idx = tid; idx < 16 * FIN; idx += 32 * WAVES) {
        int r = idx / FIN, c = idx % FIN;
        int row = rowBase + r;
        xs[r * LDSP + c] = (row < n) ? X[(size_t)row * FIN + c] : 0.0f;
    }
    __syncthreads();

    const int wave = tid >> 5;
    const int lane = tid & 31;
    const int m16  = lane & 15;     // M index (A) / N index (B, C/D)
    const int kh   = lane >> 4;     // K half-select
    const int ncol = wave * 16 + m16;

    v8f acc = {};
#pragma unroll
    for (int k0 = 0; k0 < FIN; k0 += 4) {
        const int ka = k0 + 2 * kh;
        v2f a, b;
        // A 16x4 f32: lane m, VGPR0/1 = K = 2*kh, 2*kh+1
        a.x = xs[m16 * LDSP + ka];
        a.y = xs[m16 * LDSP + ka + 1];
        // B 4x16 f32: lane n, VGPR0/1 = K rows ka, ka+1
        b.x = W[(size_t)ka * FOUT + ncol];
        b.y = W[(size_t)(ka + 1) * FOUT + ncol];
        acc = __builtin_amdgcn_wmma_f32_16x16x4_f32(
            /*neg_a=*/false, a, /*neg_b=*/false, b,
            /*c_mod=*/(short)0, acc, /*reuse_a=*/false, /*reuse_b=*/false);
    }

    // D 16x16 f32: VGPR v holds M = v + 8*kh, N = lane&15
#pragma unroll
    for (int v = 0; v < 8; ++v) {
        int row = rowBase + v + 8 * kh;
        if (row < n) Y[(size_t)row * FOUT + ncol] = acc[v];
    }
}

// ---------------- aggregation ----------------

// out[i,f] = dinv[i]^2 * h[i,f] + b[f]   (self-loop term; also initializes out)
template <int F>
__global__ void self_bias_k(const float* __restrict__ h, const float* __restrict__ dinv,
                            const float* __restrict__ b, float* __restrict__ out, int n) {
    int tid = blockIdx.x * blockDim.x + threadIdx.x;
    if (tid >= n * F) return;
    int i = tid / F, f = tid % F;
    float di = dinv[i];
    out[tid] = di * di * h[tid] + b[f];
}

// out[dst,f] += norm[e] * h[src,f]
template <int F>
__global__ void scatter_k(const int* __restrict__ src, const int* __restrict__ dst,
                          const float* __restrict__ norm, const float* __restrict__ h,
                          float* __restrict__ out, int e) {
    int tid = blockIdx.x * blockDim.x + threadIdx.x;
    if (tid >= e * F) return;
    int ed = tid / F, f = tid % F;
    float v = norm[ed] * h[(size_t)src[ed] * F + f];
    atomicAdd(&out[(size_t)dst[ed] * F + f], v);
}

// ---------------- batchnorm (training mode, biased variance) ----------------

__global__ void zero_k(float* __restrict__ p, int n) {
    int i = blockIdx.x * blockDim.x + threadIdx.x;
    if (i < n) p[i] = 0.0f;
}

template <int F>
__global__ __launch_bounds__(F)
void bn_stats_k(const float* __restrict__ h, float* __restrict__ stats, int n, int rowsPerBlock) {
    int f  = threadIdx.x;
    int r0 = blockIdx.x * rowsPerBlock;
    int r1 = r0 + rowsPerBlock;
    if (r1 > n) r1 = n;
    float s = 0.0f, s2 = 0.0f;
    for (int r = r0; r < r1; ++r) {
        float v = h[(size_t)r * F + f];
        s += v;
        s2 += v * v;
    }
    atomicAdd(&stats[f], s);
    atomicAdd(&stats[F + f], s2);
}

template <int F>
__global__ void bn_apply_relu_k(float* __restrict__ h, const float* __restrict__ stats,
                                const float* __restrict__ gamma, const float* __restrict__ beta,
                                int n) {
    int tid = blockIdx.x * blockDim.x + threadIdx.x;
    if (tid >= n * F) return;
    int f = tid % F;
    float inv_n = 1.0f / (float)n;
    float mean  = stats[f] * inv_n;
    float var   = stats[F + f] * inv_n - mean * mean;
    float w     = rsqrtf(var + EPS_BN);
    float v     = gamma[f] * (h[tid] - mean) * w + beta[f];
    h[tid] = v > 0.0f ? v : 0.0f;
}

// ---------------- driver ----------------

extern "C" void kernel_launch(void* const* d_in, const int* in_sizes, int n_in,
                              void* d_out, int out_size, void* d_ws, size_t ws_size,
                              hipStream_t stream) {
    const float* x      = (const float*)d_in[0];
    const int*   src    = (const int*)d_in[1];
    const int*   dst    = (const int*)d_in[2];
    const float* weight = (const float*)d_in[3];
    const float* W1     = (const float*)d_in[4];
    const float* b1     = (const float*)d_in[5];
    const float* g1     = (const float*)d_in[6];
    const float* be1    = (const float*)d_in[7];
    const float* W2     = (const float*)d_in[8];
    const float* b2     = (const float*)d_in[9];
    const float* g2     = (const float*)d_in[10];
    const float* be2    = (const float*)d_in[11];
    const float* W3     = (const float*)d_in[12];
    const float* b3     = (const float*)d_in[13];
    float*       out    = (float*)d_out;

    const int n = NN, e = NE;

    // workspace carve-out (256B aligned)
    size_t off = 0;
    char*  base = (char*)d_ws;
    auto take = [&](size_t bytes) -> float* {
        float* p = (float*)(base + off);
        off += (bytes + 255) & ~(size_t)255;
        return p;
    };
    float* deg   = take((size_t)n * 4);
    float* dinv  = take((size_t)n * 4);
    float* norm  = take((size_t)e * 4);
    float* h_lin = take((size_t)n * 128 * 4);  // GEMM output (max width)
    float* h_agg = take((size_t)n * 128 * 4);  // aggregated / activation buffer
    float* stats = take(256 * 4);

    const int T = 256;
    const int rowTiles = (n + 15) / 16;
    const int bnChunk  = 256;
    const int bnBlocks = (n + bnChunk - 1) / bnChunk;

    // ---- normalization precompute ----
    deg_init_k<<<(n + T - 1) / T, T, 0, stream>>>(deg, n);
    deg_edge_k<<<(e + T - 1) / T, T, 0, stream>>>(dst, weight, deg, e);
    dinv_k<<<(n + T - 1) / T, T, 0, stream>>>(deg, dinv, n);
    norm_k<<<(e + T - 1) / T, T, 0, stream>>>(src, dst, weight, dinv, norm, e);

    // ---- layer 1: 128 -> 128, BN + ReLU ----
    gemm_wmma_k<128, 128><<<rowTiles, 256, 0, stream>>>(x, W1, h_lin, n);
    self_bias_k<128><<<(n * 128 + T - 1) / T, T, 0, stream>>>(h_lin, dinv, b1, h_agg, n);
    scatter_k<128><<<(e * 128 + T - 1) / T, T, 0, stream>>>(src, dst, norm, h_lin, h_agg, e);
    zero_k<<<1, 256, 0, stream>>>(stats, 256);
    bn_stats_k<128><<<bnBlocks, 128, 0, stream>>>(h_agg, stats, n, bnChunk);
    bn_apply_relu_k<128><<<(n * 128 + T - 1) / T, T, 0, stream>>>(h_agg, stats, g1, be1, n);

    // ---- layer 2: 128 -> 64, BN + ReLU ----
    gemm_wmma_k<128, 64><<<rowTiles, 128, 0, stream>>>(h_agg, W2, h_lin, n);
    self_bias_k<64><<<(n * 64 + T - 1) / T, T, 0, stream>>>(h_lin, dinv, b2, h_agg, n);
    scatter_k<64><<<(e * 64 + T - 1) / T, T, 0, stream>>>(src, dst, norm, h_lin, h_agg, e);
    zero_k<<<1, 256, 0, stream>>>(stats, 256);
    bn_stats_k<64><<<bnBlocks, 64, 0, stream>>>(h_agg, stats, n, bnChunk);
    bn_apply_relu_k<64><<<(n * 64 + T - 1) / T, T, 0, stream>>>(h_agg, stats, g2, be2, n);

    // ---- layer 3: 64 -> 32, no BN ----
    gemm_wmma_k<64, 32><<<rowTiles, 64, 0, stream>>>(h_agg, W3, h_lin, n);
    self_bias_k<32><<<(n * 32 + T - 1) / T, T, 0, stream>>>(h_lin, dinv, b3, out, n);
    scatter_k<32><<<(e * 32 + T - 1) / T, T, 0, stream>>>(src, dst, norm, h_lin, out, e);
}